// APRConv_36653250904487
// MI455X (gfx1250) — compile-verified
//
#include <hip/hip_runtime.h>

typedef __attribute__((ext_vector_type(2))) float v2f;
typedef __attribute__((ext_vector_type(4))) float v4f;
typedef __attribute__((ext_vector_type(8))) float v8f;

constexpr int kB    = 2;
constexpr int kCin  = 16;
constexpr int kCout = 16;
constexpr int kN    = 131072;   // 2^17
constexpr int kK    = 27;
constexpr int kS    = 3;
constexpr int kWPermEntries = kS * kK * 4 * 32;   // 10368 float2 entries (== full weight tensor, permuted)

// ---------------------------------------------------------------------------
// Kernel 1: transpose x[b][c][n] -> xt[b][n][c]  (makes each gathered point a
// contiguous 64B row so the gather in the WMMA kernel is b64 loads, not 16
// strided dword loads). Reads are coalesced per-c; writes are 4x b128 per lane.
// ---------------------------------------------------------------------------
__global__ void xpose_kernel(const float* __restrict__ x, float* __restrict__ xt) {
  const unsigned gid = blockIdx.x * blockDim.x + threadIdx.x;
  const unsigned b = gid >> 17;          // kN == 2^17
  const unsigned n = gid & (kN - 1);
  float v[kCin];
#pragma unroll
  for (int c = 0; c < kCin; ++c)
    v[c] = x[((size_t)(b * kCin + c)) * kN + n];
  float* dst = xt + ((size_t)b * kN + n) * kCin;
#pragma unroll
  for (int q = 0; q < 4; ++q) {
    v4f t;
    t[0] = v[4 * q + 0]; t[1] = v[4 * q + 1];
    t[2] = v[4 * q + 2]; t[3] = v[4 * q + 3];
    *(v4f*)(dst + 4 * q) = t;
  }
}

// ---------------------------------------------------------------------------
// Kernel 2: permute weights into the exact per-lane A-operand layout of
// V_WMMA_F32_16X16X4_F32 (ISA 7.12.2, 32-bit A 16x4):
//   lane L: row M = o = L&15 ; VGPR0 = K = (L>=16 ? 2:0), VGPR1 = K+1.
// Entry e = ((s*27 + k)*4 + j)*32 + lane holds {w[s,o,c0,k], w[s,o,c0+1,k]}
// with c0 = 4*j + 2*(lane>>4). One coalesced b64 load per wave per (s,k,j).
// ---------------------------------------------------------------------------
__global__ void wperm_kernel(const float* __restrict__ w, v2f* __restrict__ wp) {
  const int e = blockIdx.x * blockDim.x + threadIdx.x;
  if (e >= kWPermEntries) return;
  const int lane = e & 31;
  const int j    = (e >> 5) & 3;
  const int sk   = e >> 7;
  const int k    = sk % kK;
  const int s    = sk / kK;
  const int o    = lane & 15;
  const int c0   = 4 * j + ((lane >> 4) << 1);
  const float* src = w + ((size_t)(s * kCout + o) * kCin + c0) * kK + k;
  v2f a;
  a[0] = src[0];
  a[1] = src[kK];     // c0+1 is kK floats away in [s][o][c][k] layout
  wp[e] = a;
}

// ---------------------------------------------------------------------------
// Main kernel: one wave32 per 16-point tile. D = W_s (16x432) * X (432x16) via
// 108 chained V_WMMA_F32_16X16X4_F32 per (b,s). All 3 stencils computed, final
// per-point select + bias.
// ---------------------------------------------------------------------------
template <bool FAST>
__global__ void aprconv_wmma_kernel(const float* __restrict__ xt,    // FAST: [b][n][c]
                                    const float* __restrict__ x,     // raw   [b][c][n]
                                    const v2f*  __restrict__ wperm,  // FAST
                                    const float* __restrict__ w,     // raw
                                    const float* __restrict__ bias,
                                    const int*  __restrict__ nbr,    // [n][k]
                                    const int*  __restrict__ sten,   // [n]
                                    float* __restrict__ out) {       // [b][o][n]
  const int lane = threadIdx.x & 31;
  const int wave = threadIdx.x >> 5;
  const int tile = blockIdx.x * (blockDim.x >> 5) + wave;
  const int n0   = tile << 4;
  const int col  = lane & 15;        // point column / weight row (o)
  const int hi   = lane >> 4;        // which K-half of the x4 step
  const int crel = hi << 1;          // 0 or 2

  const v8f vzero = {0.f, 0.f, 0.f, 0.f, 0.f, 0.f, 0.f, 0.f};
  v8f acc[kB][kS];
#pragma unroll
  for (int b = 0; b < kB; ++b)
#pragma unroll
    for (int s = 0; s < kS; ++s) acc[b][s] = vzero;

  const int* nbr_col = nbr + (size_t)(n0 + col) * kK;

  for (int k = 0; k < kK; ++k) {
    const int idx = nbr_col[k];      // lanes L and L+16 fetch same column index

    // --- B operand: X_k[c][col], 4 chunks of K=4 along c, both batches ------
    v2f bm[kB][4];
#pragma unroll
    for (int b = 0; b < kB; ++b) {
      if (FAST) {
        const float* src = xt + (((size_t)b * kN + idx) << 4) + crel;
#pragma unroll
        for (int j = 0; j < 4; ++j)
          bm[b][j] = *(const v2f*)(src + 4 * j);       // b64 gather, 8B aligned
      } else {
        const float* src = x + (size_t)b * kCin * kN + idx;
#pragma unroll
        for (int j = 0; j < 4; ++j) {
          v2f t;
          t[0] = src[(size_t)(4 * j + crel) * kN];
          t[1] = src[(size_t)(4 * j + crel + 1) * kN];
          bm[b][j] = t;
        }
      }
    }

    // --- A operand (weights) loaded once, reused for both batches -----------
#pragma unroll
    for (int s = 0; s < kS; ++s) {
#pragma unroll
      for (int j = 0; j < 4; ++j) {
        v2f a;
        if (FAST) {
          a = wperm[(((s * kK + k) * 4 + j) << 5) + lane];   // coalesced b64, L2-hit
        } else {
          const int c0 = 4 * j + crel;
          const float* wpp = w + ((size_t)(s * kCout + col) * kCin + c0) * kK + k;
          a[0] = wpp[0];
          a[1] = wpp[kK];
        }
#pragma unroll
        for (int b = 0; b < kB; ++b)
          acc[b][s] = __builtin_amdgcn_wmma_f32_16x16x4_f32(
              false, a, false, bm[b][j], (short)0, acc[b][s], false, false);
      }
    }
  }

  // --- epilogue: select per-point stencil, add bias, store -------------------
  // C/D layout: VGPR v, lane L -> M = v + 8*(L>>4) (= o), N = L&15 (= col)
  const int n   = n0 + col;
  const int sid = sten[n];
  float bv[8];
#pragma unroll
  for (int v = 0; v < 8; ++v) bv[v] = bias[hi * 8 + v];
#pragma unroll
  for (int b = 0; b < kB; ++b) {
    v8f r = (sid == 0) ? acc[b][0] : ((sid == 1) ? acc[b][1] : acc[b][2]);
#pragma unroll
    for (int v = 0; v < 8; ++v)
      out[((size_t)(b * kCout + hi * 8 + v)) * kN + n] = r[v] + bv[v];
  }
}

// ---------------------------------------------------------------------------
extern "C" void kernel_launch(void* const* d_in, const int* in_sizes, int n_in,
                              void* d_out, int out_size, void* d_ws, size_t ws_size,
                              hipStream_t stream) {
  const float* x    = (const float*)d_in[0];
  const float* w    = (const float*)d_in[1];
  const float* bias = (const float*)d_in[2];
  const int*   nbr  = (const int*)d_in[3];
  const int*   sten = (const int*)d_in[4];
  float* out = (float*)d_out;

  const size_t xt_bytes = (size_t)kB * kN * kCin * sizeof(float);   // 16 MiB
  const size_t wp_bytes = (size_t)kWPermEntries * sizeof(v2f);      // 81 KiB
  const size_t need     = xt_bytes + wp_bytes;

  const int tiles = kN / 16;           // 8192 tiles, one wave32 each
  dim3 block(256);                     // 8 waves / block
  dim3 grid(tiles / 8);                // 1024 blocks

  if (ws_size >= need) {
    float* xt   = (float*)d_ws;
    v2f* wperm  = (v2f*)((char*)d_ws + xt_bytes);
    xpose_kernel<<<(kB * kN) / 256, 256, 0, stream>>>(x, xt);
    wperm_kernel<<<(kWPermEntries + 255) / 256, 256, 0, stream>>>(w, wperm);
    aprconv_wmma_kernel<true><<<grid, block, 0, stream>>>(xt, x, wperm, w, bias, nbr, sten, out);
  } else {
    aprconv_wmma_kernel<false><<<grid, block, 0, stream>>>(nullptr, x, nullptr, w, bias, nbr, sten, out);
  }
}